// SymmetricContraction_30691836297234
// MI455X (gfx1250) — compile-verified
//
#include <hip/hip_runtime.h>

typedef __attribute__((ext_vector_type(2))) float v2f;
typedef __attribute__((ext_vector_type(8))) float v8f;

#define MULC   128   // channels
#define LM     9     // irrep length (1+3+5)
#define KP     12    // K padded to multiple of 4
#define NCOMP  4     // 1 scalar + 3 vector outputs
#define PR     96    // rows per component padded 81 -> 96
#define NTILE  6     // 96 / 16
#define ZS     20    // Z row stride in floats (16 + 4 pad, kills store bank conflicts)
#define WAVES  4
#define BT     64    // batch elements per workgroup (16 per wave)

__global__ __launch_bounds__(128)
void symcon_wmma_kernel(const float* __restrict__ x,
                        const float* __restrict__ U3s, const float* __restrict__ U2s,
                        const float* __restrict__ U1s, const float* __restrict__ Wms,
                        const float* __restrict__ W2s, const float* __restrict__ W1s,
                        const float* __restrict__ U3v, const float* __restrict__ U2v,
                        const float* __restrict__ U1v, const float* __restrict__ Wmv,
                        const float* __restrict__ W2v, const float* __restrict__ W1v,
                        float* __restrict__ out)
{
    __shared__ float sCT3[NCOMP * PR * KP];   // per-channel cubic coeffs, K-padded
    __shared__ float sC2 [NCOMP * PR];        // quadratic coeffs
    __shared__ float sC1 [NCOMP * 16];        // linear coeffs
    __shared__ float sX  [WAVES][16][KP];     // x block per wave, K-padded
    __shared__ float sZ  [WAVES][PR * ZS];    // WMMA output scratch per wave

    const int t  = threadIdx.x;
    const int c  = blockIdx.x;                // channel
    const int b0 = blockIdx.y * BT;

    // ---------------- Phase A: fold U-tensors with per-channel weights ----------------
    for (int r = t; r < NCOMP * PR; r += 128) {
        const int comp  = r / PR;
        const int row   = r % PR;             // row = p*9 + q (valid when < 81)
        const bool valid = row < 81;
        float c2 = 0.f;
        if (valid) {
            if (comp == 0) {
                c2 = U2s[row*2+0]*W2s[0*MULC+c] + U2s[row*2+1]*W2s[1*MULC+c];
            } else {
                const int m = comp - 1;
                c2 = U2v[(m*81+row)*2+0]*W2v[0*MULC+c] + U2v[(m*81+row)*2+1]*W2v[1*MULC+c];
            }
        }
        sC2[r] = c2;
        for (int i = 0; i < KP; ++i) {
            float v = 0.f;
            if (valid && i < LM) {
                if (comp == 0) {
                    const float* u = &U3s[(row*LM + i)*4];
                    v = u[0]*Wms[0*MULC+c] + u[1]*Wms[1*MULC+c]
                      + u[2]*Wms[2*MULC+c] + u[3]*Wms[3*MULC+c];
                } else {
                    const int m = comp - 1;
                    const float* u = &U3v[((m*81 + row)*LM + i)*5];
                    v = u[0]*Wmv[0*MULC+c] + u[1]*Wmv[1*MULC+c] + u[2]*Wmv[2*MULC+c]
                      + u[3]*Wmv[3*MULC+c] + u[4]*Wmv[4*MULC+c];
                }
            }
            sCT3[r*KP + i] = v;
        }
    }
    for (int r = t; r < NCOMP * 16; r += 128) {
        const int comp = r >> 4;
        const int p    = r & 15;
        float v = 0.f;
        if (p < LM) v = (comp == 0) ? U1s[p] * W1s[c] : U1v[(comp-1)*LM + p] * W1v[c];
        sC1[r] = v;
    }

    // ---------------- Stage x block (16 b per wave) into LDS ----------------
    const int wave  = t >> 5;
    const int lane  = t & 31;
    const int n     = lane & 15;              // local batch column / matrix N index
    const int half  = lane >> 4;              // lane half selects K sub-pair in fragments
    const int bbase = b0 + wave * 16;

    for (int s = lane; s < 16 * KP; s += 32) {
        const int bb = s / KP;
        const int i  = s % KP;
        float v = 0.f;
        if (i < LM) {
            const size_t gb = (size_t)(bbase + bb) * 1152;
            if (i == 0)      v = x[gb + c];                      // 0e block
            else if (i < 4)  v = x[gb + 128 + c*3 + (i - 1)];    // 1o block
            else             v = x[gb + 512 + c*5 + (i - 4)];    // 2e block
        }
        sX[wave][bb][i] = v;
    }
    __syncthreads();

    float xv[LM];
    #pragma unroll
    for (int i = 0; i < LM; ++i) xv[i] = sX[wave][n][i];

    const int koff = half << 1;               // 0 for lanes 0-15, 2 for lanes 16-31
    float* Zw = &sZ[wave][0];
    const float* xrow = &sX[wave][n][0];

    for (int comp = 0; comp < NCOMP; ++comp) {
        // ---- WMMA GEMM: Z[row, n] = sum_i T3[comp,row,i] * x[i, n] ----
        #pragma unroll
        for (int tile = 0; tile < NTILE; ++tile) {
            v8f acc = {0.f, 0.f, 0.f, 0.f, 0.f, 0.f, 0.f, 0.f};
            const float* crow = &sCT3[((comp * PR) + tile * 16 + n) * KP];
            #pragma unroll
            for (int k0 = 0; k0 < KP; k0 += 4) {
                v2f a  = { crow[k0 + koff], crow[k0 + koff + 1] };   // A: M=row, K pair
                v2f bv = { xrow[k0 + koff], xrow[k0 + koff + 1] };   // B: K pair, N=n
                acc = __builtin_amdgcn_wmma_f32_16x16x4_f32(
                          false, a, false, bv, (short)0, acc, false, false);
            }
            const int rb = tile * 16 + (half << 3);
            #pragma unroll
            for (int r = 0; r < 8; ++r)
                Zw[(rb + r) * ZS + n] = acc[r];
        }
        __syncthreads();

        // ---- post: Σ_{p,q} (Z + T2)·x_p·x_q  +  Σ_p T1·x_p ----
        float accq = 0.f;
        for (int p = 0; p < LM; ++p) {
            const float xp_ = xv[p];
            for (int q = half; q < LM; q += 2) {      // q parity split across lane halves
                const int rr = p * LM + q;
                accq += (Zw[rr * ZS + n] + sC2[comp * PR + rr]) * (xp_ * xv[q]);
            }
        }
        float lin = 0.f;
        #pragma unroll
        for (int p = 0; p < LM; ++p) lin += sC1[comp * 16 + p] * xv[p];

        accq += __shfl_xor(accq, 16, 32);             // combine the two halves (wave32)
        if (half == 0) {
            const int col = (comp == 0) ? c : (128 + c * 3 + (comp - 1));
            out[(size_t)(bbase + n) * 512 + col] = accq + lin;
        }
        __syncthreads();                              // protect Z before next component
    }
}

extern "C" void kernel_launch(void* const* d_in, const int* in_sizes, int n_in,
                              void* d_out, int out_size, void* d_ws, size_t ws_size,
                              hipStream_t stream)
{
    (void)n_in; (void)out_size; (void)d_ws; (void)ws_size;
    const float* x   = (const float*)d_in[0];
    const float* U3s = (const float*)d_in[1];
    const float* U2s = (const float*)d_in[2];
    const float* U1s = (const float*)d_in[3];
    const float* Wms = (const float*)d_in[4];
    const float* W2s = (const float*)d_in[5];
    const float* W1s = (const float*)d_in[6];
    const float* U3v = (const float*)d_in[7];
    const float* U2v = (const float*)d_in[8];
    const float* U1v = (const float*)d_in[9];
    const float* Wmv = (const float*)d_in[10];
    const float* W2v = (const float*)d_in[11];
    const float* W1v = (const float*)d_in[12];
    float* out = (float*)d_out;

    const int nB = in_sizes[0] / 1152;        // 2048 for the reference shapes
    dim3 grid(MULC, nB / BT);
    symcon_wmma_kernel<<<grid, 128, 0, stream>>>(
        x, U3s, U2s, U1s, Wms, W2s, W1s,
        U3v, U2v, U1v, Wmv, W2v, W1v, out);
}